// SteerableFirstLayerPerSample_31559419691731
// MI455X (gfx1250) — compile-verified
//
#include <hip/hip_runtime.h>
#include <hip/hip_bf16.h>
#include <math.h>

#ifndef M_PI
#define M_PI 3.14159265358979323846
#endif

#define AS1 __attribute__((address_space(1)))
#define AS3 __attribute__((address_space(3)))

typedef __attribute__((ext_vector_type(2))) float v2f;
typedef __attribute__((ext_vector_type(8))) float v8f;
typedef __attribute__((ext_vector_type(4))) int   v4i;

// Problem constants (from reference)
constexpr int   B_  = 64;
constexpr int   IC_ = 2;
constexpr int   OC_ = 32;
constexpr int   K_  = 5;
constexpr long  L_  = 65536;
constexpr float FS_ = 50000000.0f;
constexpr int   JMAX_ = 16;

// Conv tiling
constexpr int TILE_L = 2048;              // positions per block
constexpr int HALO   = 4;                 // 16B-aligned halo (need 2, use 4)
constexpr int LSTR   = TILE_L + 2 * HALO; // LDS row stride in floats

// ---------------------------------------------------------------------------
// Kernel 1: build per-sample filters w[b][oc][ic][k] into workspace.
// Tiny: 64 blocks x 320 threads, one thread per (oc,ic,k).
// ---------------------------------------------------------------------------
__global__ void prep_weights_kernel(const int* __restrict__ z,
                                    const float* __restrict__ s,
                                    const float* __restrict__ w0,
                                    float* __restrict__ w)
{
  const int b = blockIdx.x;
  const int t = threadIdx.x;
  if (t >= OC_ * IC_ * K_) return;
  const int oc  = t / (IC_ * K_);
  const int idx = t % (IC_ * K_);
  const int ic  = idx / K_;
  const int kx  = idx % K_;

  auto param = [&](int col, float neutral) -> float {
    const int   zz = z[b * 5 + col];
    float       sv = s[b * 5 + col];
    if (isnan(sv)) sv = 0.0f;                                   // nan_to_num
    if (isinf(sv)) sv = (sv > 0.0f) ? 3.4028235e38f : -3.4028235e38f;
    return (zz > 0) ? sv : neutral;
  };
  const float alpha = param(1, 1.0f);
  const float f0    = param(0, 0.0f);
  const float aacc  = param(4, 0.0f);
  const float rho   = param(2, 1.0f);

  // new_W = clip(round(K/max(alpha,1e-3)), 1, 16)  (round half-even)
  const float r  = rintf((float)K_ / fmaxf(alpha, 0.001f));
  const int   nW = (int)fminf(fmaxf(r, 1.0f), (float)JMAX_);
  const float nWf = (float)nW;

  // Row kx of M = W2 @ W1 (W2 row has only 2 nonzeros at j0/j1)
  const float src2 = fmaxf(((float)kx + 0.5f) * nWf / (float)K_ - 0.5f, 0.0f);
  const int   j0   = (int)floorf(src2);
  const int   j1   = min(j0 + 1, nW - 1);
  const float lam2 = src2 - (float)j0;

  float Mrow[K_] = {0.f, 0.f, 0.f, 0.f, 0.f};
  {
    const int   js[2] = {j0, j1};
    const float wg[2] = {1.0f - lam2, lam2};
    for (int q = 0; q < 2; ++q) {
      const int j = js[q];
      if (j >= nW) continue;                                    // valid mask
      const float src1 = fmaxf(((float)j + 0.5f) * (float)K_ / nWf - 0.5f, 0.0f);
      const int   i0   = (int)floorf(src1);
      const int   i1   = min(i0 + 1, K_ - 1);
      const float lam1 = src1 - (float)i0;
      Mrow[i0] += wg[q] * (1.0f - lam1);
      Mrow[i1] += wg[q] * lam1;
    }
  }

  // Resample both I and Q taps (rotation mixes them)
  float wI = 0.0f, wQ = 0.0f;
  for (int y = 0; y < K_; ++y) {
    wI += Mrow[y] * w0[(oc * IC_ + 0) * K_ + y];
    wQ += Mrow[y] * w0[(oc * IC_ + 1) * K_ + y];
  }

  // Composed rotation: phi = 2*pi*f0*n/FS + pi*a*(n/FS)^2 ; theta = 0
  const float nn  = (float)kx;
  const float tr  = nn / FS_;
  const float phi = 2.0f * (float)M_PI * f0 * nn / FS_ + (float)M_PI * aacc * tr * tr;
  const float c = cosf(phi), sn = sinf(phi);
  const float out = (ic == 0) ? rho * (wI * c - wQ * sn)
                              : rho * (wI * sn + wQ * c);
  w[((size_t)(b * OC_ + oc) * IC_ + ic) * K_ + kx] = out;
}

// ---------------------------------------------------------------------------
// Kernel 2: bandwidth-bound grouped conv via WMMA f32 16x16x4.
//   GEMM view per sample: D[oc][n] += A[oc][kk] * Bm[kk][n],
//   kk = ic*5+k (10 real rows, padded to 12 = 3 WMMA K-steps),
//   Bm[kk][n] = x[ic][l0+n+k-2] gathered from an LDS-staged x tile.
// x tile is staged with CDNA5 async global->LDS B128 loads (ASYNCcnt).
// ---------------------------------------------------------------------------
__global__ __launch_bounds__(256) void conv_wmma_kernel(
    const float* __restrict__ x,   // [B][IC][L]
    const float* __restrict__ w,   // [B][OC][IC][K]
    float* __restrict__ y)         // [B][OC][L]
{
  __shared__ __align__(16) float xs[IC_][LSTR];

  const int  b    = blockIdx.y;
  const long l0   = (long)blockIdx.x * TILE_L;
  const int  tid  = threadIdx.x;
  const int  lane = tid & 31;
  const int  wv   = tid >> 5;          // wave id 0..7
  const int  n    = lane & 15;         // N (position) / A-row index
  const int  hi   = lane >> 4;         // upper lane half?

  // ---- stage x tile (bulk, 16B aligned both sides) ----
  for (int ch = 0; ch < IC_; ++ch) {
    const float* g  = x + ((size_t)b * IC_ + ch) * L_ + l0;
    float*       lp = &xs[ch][HALO];
    for (int t = tid; t < TILE_L / 4; t += 256) {
#if defined(__gfx1250__) && __has_builtin(__builtin_amdgcn_global_load_async_to_lds_b128)
      __builtin_amdgcn_global_load_async_to_lds_b128(
          (AS1 v4i*)(g + 4 * t), (AS3 v4i*)(lp + 4 * t),
          /*offset=*/0, /*cpol=*/0);
#else
      const float4 v = *(const float4*)(g + 4 * t);
      *(float4*)(lp + 4 * t) = v;
#endif
    }
  }
  // ---- halo (zero-padded at tensor edges), 16 threads ----
  if (tid < 16) {
    const int  side = tid >> 3;          // 0=left, 1=right
    const int  ch   = (tid >> 2) & 1;
    const int  j    = tid & 3;
    const long gidx = side ? (l0 + TILE_L + j) : (l0 - HALO + j);
    const float v = (gidx >= 0 && gidx < L_)
                        ? x[((size_t)b * IC_ + ch) * L_ + gidx] : 0.0f;
    xs[ch][side ? (HALO + TILE_L + j) : j] = v;
  }
#if defined(__gfx1250__)
#if __has_builtin(__builtin_amdgcn_s_wait_asynccnt)
  __builtin_amdgcn_s_wait_asynccnt(0);
#else
  asm volatile("s_wait_asynccnt 0" ::: "memory");
#endif
#endif
  __syncthreads();

  // ---- A matrices (per-sample filters), loop invariant ----
  // A 16x4 f32 layout: lane l -> row l&15; vgpr0 holds K={0|2}, vgpr1 K={1|3}
  v2f A[3][2];
  for (int h = 0; h < 2; ++h) {
    const float* Wp = w + ((size_t)b * OC_ + h * 16 + n) * (IC_ * K_); // 10 taps
    for (int c = 0; c < 3; ++c) {
      const int kk = c * 4 + (hi ? 2 : 0);
      A[c][h].x = (kk     < IC_ * K_) ? Wp[kk]     : 0.0f;  // pad kk=10,11
      A[c][h].y = (kk + 1 < IC_ * K_) ? Wp[kk + 1] : 0.0f;
    }
  }

  // ---- per-lane LDS offsets for B gathers (exclude running position) ----
  int offB0[3], offB1[3];
  for (int c = 0; c < 3; ++c) {
    int kk0 = c * 4 + (hi ? 2 : 0);
    int kk1 = kk0 + 1;
    kk0 = (kk0 > 9) ? 9 : kk0;          // padded rows: A==0, read in-range
    kk1 = (kk1 > 9) ? 9 : kk1;
    offB0[c] = (kk0 / K_) * LSTR + (kk0 % K_) - 2 + HALO + n;
    offB1[c] = (kk1 / K_) * LSTR + (kk1 % K_) - 2 + HALO + n;
  }

  const float* xf = &xs[0][0];
  float*       yb = y + (size_t)b * OC_ * L_ + l0;

  // Each wave: 256 contiguous positions, 16 at a time; 2 M-tiles (32 oc).
  for (int i = 0; i < TILE_L / (8 * 16); ++i) {
    const int tpos = wv * 256 + i * 16;
    v8f acc0 = {0.f, 0.f, 0.f, 0.f, 0.f, 0.f, 0.f, 0.f};
    v8f acc1 = {0.f, 0.f, 0.f, 0.f, 0.f, 0.f, 0.f, 0.f};
    for (int c = 0; c < 3; ++c) {
      v2f Bv;
      Bv.x = xf[offB0[c] + tpos];
      Bv.y = xf[offB1[c] + tpos];
      acc0 = __builtin_amdgcn_wmma_f32_16x16x4_f32(
          false, A[c][0], false, Bv, (short)0, acc0, false, false);
      acc1 = __builtin_amdgcn_wmma_f32_16x16x4_f32(
          false, A[c][1], false, Bv, (short)0, acc1, false, false);
    }
    // C/D layout: vgpr r -> M=r (lanes 0-15) / M=r+8 (lanes 16-31), N=lane&15
    for (int r = 0; r < 8; ++r) {
      const int ocr = r + hi * 8;
      yb[(size_t)ocr        * L_ + tpos + n] = acc0[r];
      yb[(size_t)(16 + ocr) * L_ + tpos + n] = acc1[r];
    }
  }
}

// ---------------------------------------------------------------------------
extern "C" void kernel_launch(void* const* d_in, const int* in_sizes, int n_in,
                              void* d_out, int out_size, void* d_ws, size_t ws_size,
                              hipStream_t stream) {
  const float* x  = (const float*)d_in[0];
  const int*   z  = (const int*)d_in[1];
  const float* s  = (const float*)d_in[2];
  const float* w0 = (const float*)d_in[3];
  float* y = (float*)d_out;
  float* w = (float*)d_ws;   // B*OC*IC*K = 20480 floats (80 KB) scratch

  prep_weights_kernel<<<B_, 320, 0, stream>>>(z, s, w0, w);

  dim3 grid((unsigned)(L_ / TILE_L), (unsigned)B_);  // 32 x 64 blocks
  conv_wmma_kernel<<<grid, 256, 0, stream>>>(x, w, y);
}